// HardLSTM2_12876311954122
// MI455X (gfx1250) — compile-verified
//
#include <hip/hip_runtime.h>

// Problem constants (match reference)
#define T_DIM 512
#define B_DIM 64
#define IN_DIM 512
#define H_DIM 512
#define G_DIM 2048   // 4*H

// LDS panel row stride (floats). 72 => banks for lanes 0-15 (row k) and
// lanes 16-31 (row k+2) occupy disjoint 16-bank windows: 2*72 % 64 == 16.
// TDM reproduces this stride via pad_interval=16 dwords, pad_amount=56 dwords.
#define PANEL_STRIDE 72
#define NBLOCKS_SEQ 64

typedef __attribute__((ext_vector_type(2))) float    v2f;
typedef __attribute__((ext_vector_type(8))) float    v8f;
typedef __attribute__((ext_vector_type(4))) unsigned v4u;
typedef __attribute__((ext_vector_type(8))) int      v8i;
typedef __attribute__((ext_vector_type(4))) int      v4i;

__device__ __forceinline__ float hsig(float x) {
    return fminf(fmaxf(0.2f * x + 0.5f, 0.0f), 1.0f);
}
__device__ __forceinline__ float clip1(float x) {
    return fminf(fmaxf(x, -1.0f), 1.0f);
}

// ---------------------------------------------------------------------------
// TDM: stage one gate panel (512 rows x 16 cols, row stride G dwords in
// global) into LDS with 72-dword row stride via the pad feature.
// D# layout per CDNA5 ISA ch.8 (groups 0/1; groups 2/3 zero for 2D tensors).
// This toolchain exposes the 6-arg builtin: (v4u, v8i, v4i, v4i, v8i, cpol).
// ---------------------------------------------------------------------------
__device__ __forceinline__ void tdm_load_gate_panel(const float* gsrc,
                                                    unsigned lds_byte_addr) {
    unsigned long long ga = (unsigned long long)(uintptr_t)gsrc;
    v4u g0;
    g0[0] = 1u;                                   // count=1 (valid), user mode
    g0[1] = lds_byte_addr;                        // lds_addr
    g0[2] = (unsigned)ga;                         // global_addr[31:0]
    g0[3] = (unsigned)((ga >> 32) & 0x1FFFFFFull) // global_addr[56:32]
            | (2u << 30);                         // type=2 ("image")
    v8i g1;
    g1[0] = (int)((2u << 16)     // data_size = 4 bytes
                | (1u << 20)     // pad_enable
                | (3u << 22)     // pad_interval: 16 dwords
                | (55u << 25));  // pad_amount: 56 dwords  -> 72-dword rows
    g1[1] = (int)(16u << 16);    // tensor_dim0 = 16 (lo16 @ bits 48..63)
    g1[2] = (int)(512u << 16);   // tensor_dim0 hi=0 | tensor_dim1 = 512 (lo16)
    g1[3] = (int)(16u << 16);    // tensor_dim1 hi=0 | tile_dim0 = 16
    g1[4] = (int)512;            // tile_dim1 = 512 | tile_dim2 = 0
    g1[5] = (int)G_DIM;          // tensor_dim0_stride = 2048 (lo32)
    g1[6] = 0;                   // stride hi | tensor_dim1_stride lo
    g1[7] = 0;
    v4i z4 = {0, 0, 0, 0};
    v8i z8 = {0, 0, 0, 0, 0, 0, 0, 0};
    __builtin_amdgcn_tensor_load_to_lds(g0, g1, z4, z4, z8, 0);
}

// ---------------------------------------------------------------------------
// Prep: transpose weights [G,K] -> [K,G] for coalesced B loads, fold biases,
// and zero the device-wide barrier counter.
// ---------------------------------------------------------------------------
__global__ void prep_kernel(const float* __restrict__ wih_f, const float* __restrict__ whh_f,
                            const float* __restrict__ bih_f, const float* __restrict__ bhh_f,
                            const float* __restrict__ wih_b, const float* __restrict__ whh_b,
                            const float* __restrict__ bih_b, const float* __restrict__ bhh_b,
                            float* __restrict__ wihT_f, float* __restrict__ whhT_f,
                            float* __restrict__ wihT_b, float* __restrict__ whhT_b,
                            float* __restrict__ bias_f, float* __restrict__ bias_b,
                            unsigned* __restrict__ bar) {
    int idx = blockIdx.x * blockDim.x + threadIdx.x;
    if (idx < G_DIM * IN_DIM) {
        int g = idx / IN_DIM;
        int k = idx % IN_DIM;
        int t = k * G_DIM + g;
        wihT_f[t] = wih_f[idx];
        whhT_f[t] = whh_f[idx];
        wihT_b[t] = wih_b[idx];
        whhT_b[t] = whh_b[idx];
    }
    if (idx < G_DIM) {
        bias_f[idx] = bih_f[idx] + bhh_f[idx];
        bias_b[idx] = bih_b[idx] + bhh_b[idx];
    }
    if (idx == 0) *bar = 0u;
}

// ---------------------------------------------------------------------------
// Input-gate GEMM: xg[dir][t*B+b][g] = sum_k x[t,b,k] * wT[k][g] + bias[g]
// One wave per 16-row x 64-col strip: 4 accumulators, A fragment reused 4x.
// FP32 WMMA 16x16x4 (exact FP32 math, matches reference).
// ---------------------------------------------------------------------------
__global__ void xg_gemm_kernel(const float* __restrict__ x,
                               const float* __restrict__ wihT_f,
                               const float* __restrict__ wihT_b,
                               const float* __restrict__ bias_f,
                               const float* __restrict__ bias_b,
                               float* __restrict__ xg_f,
                               float* __restrict__ xg_b) {
    const int lane = threadIdx.x & 31;
    const int wgid = (blockIdx.x * blockDim.x + threadIdx.x) >> 5;
    const int NG = G_DIM / 64;             // 32 column groups of 64
    const int MT = (T_DIM * B_DIM) / 16;   // 2048 row tiles
    const int dir   = wgid / (MT * NG);
    const int rem   = wgid % (MT * NG);
    const int mtile = rem / NG;
    const int ngrp  = rem % NG;

    const float* __restrict__ wT   = dir ? wihT_b : wihT_f;
    const float* __restrict__ bias = dir ? bias_b : bias_f;
    float* __restrict__ xg         = dir ? xg_b   : xg_f;

    const int mrow = lane & 15;            // A: M index / B: N index
    const int kofs = (lane >> 4) * 2;      // lanes 16-31 hold K+2,K+3
    const int gbase = ngrp * 64;

    const float* __restrict__ aptr = x + (size_t)(mtile * 16 + mrow) * IN_DIM;

    v8f acc0 = {}, acc1 = {}, acc2 = {}, acc3 = {};
    for (int k0 = 0; k0 < IN_DIM; k0 += 4) {
        v2f a;
        a.x = aptr[k0 + kofs];
        a.y = aptr[k0 + kofs + 1];
        const float* __restrict__ b0 = wT + (size_t)(k0 + kofs) * G_DIM + gbase + mrow;
        const float* __restrict__ b1 = b0 + G_DIM;
        v2f bb;
        bb.x = b0[0];  bb.y = b1[0];
        acc0 = __builtin_amdgcn_wmma_f32_16x16x4_f32(false, a, false, bb, (short)0, acc0, false, false);
        bb.x = b0[16]; bb.y = b1[16];
        acc1 = __builtin_amdgcn_wmma_f32_16x16x4_f32(false, a, false, bb, (short)0, acc1, false, false);
        bb.x = b0[32]; bb.y = b1[32];
        acc2 = __builtin_amdgcn_wmma_f32_16x16x4_f32(false, a, false, bb, (short)0, acc2, false, false);
        bb.x = b0[48]; bb.y = b1[48];
        acc3 = __builtin_amdgcn_wmma_f32_16x16x4_f32(false, a, false, bb, (short)0, acc3, false, false);
    }

    // C/D layout: VGPR r -> M=r (lanes 0-15) or M=r+8 (lanes 16-31), N=lane&15
    const int n   = lane & 15;
    const int mhi = (lane >> 4) * 8;
    for (int r = 0; r < 8; ++r) {
        size_t row = (size_t)(mtile * 16 + r + mhi);
        int col = gbase + n;
        xg[row * G_DIM + col]      = acc0[r] + bias[col];
        xg[row * G_DIM + col + 16] = acc1[r] + bias[col + 16];
        xg[row * G_DIM + col + 32] = acc2[r] + bias[col + 32];
        xg[row * G_DIM + col + 48] = acc3[r] + bias[col + 48];
    }
}

// ---------------------------------------------------------------------------
// Persistent recurrence kernel: 64 blocks x 128 threads = 256 waves, one wave
// per (dir, btile, htile) tile, all 512 timesteps in one launch.
//  - The 512x64 w_hh gate panel for this (dir, htile) is staged into LDS once
//    via four TDM descriptors (pad feature gives the 72-dword row stride ->
//    conflict-free ds_load_b32 B-fragments) and reused by every timestep.
//  - Cell state lives in 8 registers per lane for the whole sequence.
//  - Steps are separated by an agent-scope atomic arrival counter barrier.
// ---------------------------------------------------------------------------
__global__ void lstm_seq_kernel(const float* __restrict__ h0,     // [2,B,H]
                                const float* __restrict__ c0,     // [2,B,H]
                                const float* __restrict__ whhT_f, // [H,G]
                                const float* __restrict__ whhT_b,
                                const float* __restrict__ xg_f,   // [T,B,G]
                                const float* __restrict__ xg_b,
                                float* __restrict__ out,          // [T,B,2H] then hy,cy
                                unsigned* __restrict__ bar) {
    extern __shared__ float panel[];   // [512][PANEL_STRIDE]

    const int tid   = threadIdx.x;
    const int lane  = tid & 31;
    const int wave  = tid >> 5;          // btile 0..3
    const int dir   = blockIdx.x >> 5;
    const int htile = blockIdx.x & 31;   // 0..31

    const float* __restrict__ whhT = dir ? whhT_b : whhT_f;
    const float* __restrict__ xg   = dir ? xg_b   : xg_f;

    // Stage the 4-gate B panel with the Tensor Data Mover (wave 0 issues; TDM
    // ignores EXEC so the uniform branch yields one issue per descriptor).
    // Result: panel[k][g*16+n] = whhT[k][g*H + htile*16 + n], rows 72 dwords.
    if (wave == 0) {
        unsigned lbase = (unsigned)(uintptr_t)panel;
        for (int g = 0; g < 4; ++g) {
            tdm_load_gate_panel(whhT + (size_t)g * H_DIM + htile * 16,
                                lbase + (unsigned)g * 64u);
        }
        __builtin_amdgcn_s_wait_tensorcnt(0);
    }
    __syncthreads();

    const int mrow = lane & 15;
    const int kofs = (lane >> 4) * 2;
    const int mhi  = (lane >> 4) * 8;
    const int hcol = htile * 16 + mrow;
    const int brow = wave * 16 + mrow;   // batch row this lane's A-fragment uses

    // Register-resident cell state: lane owns (b = wave*16 + r + mhi, hcol)
    float creg[8];
    for (int r = 0; r < 8; ++r)
        creg[r] = c0[(size_t)dir * B_DIM * H_DIM
                     + (size_t)(wave * 16 + r + mhi) * H_DIM + hcol];

    for (int s = 0; s < T_DIM; ++s) {
        const float* __restrict__ hp = (s == 0)
            ? h0 + (size_t)dir * B_DIM * H_DIM + (size_t)brow * H_DIM
            : out + (size_t)(s - 1) * B_DIM * 2 * H_DIM
                  + (size_t)brow * 2 * H_DIM + (size_t)dir * H_DIM;

        v8f acc0 = {}, acc1 = {}, acc2 = {}, acc3 = {};
        for (int k0 = 0; k0 < H_DIM; k0 += 4) {
            v2f a;
            a.x = hp[k0 + kofs];
            a.y = hp[k0 + kofs + 1];
            const float* __restrict__ l0 = panel + (k0 + kofs) * PANEL_STRIDE + mrow;
            const float* __restrict__ l1 = l0 + PANEL_STRIDE;
            v2f bb;
            bb.x = l0[0];  bb.y = l1[0];     // gate i
            acc0 = __builtin_amdgcn_wmma_f32_16x16x4_f32(false, a, false, bb, (short)0, acc0, false, false);
            bb.x = l0[16]; bb.y = l1[16];    // gate f
            acc1 = __builtin_amdgcn_wmma_f32_16x16x4_f32(false, a, false, bb, (short)0, acc1, false, false);
            bb.x = l0[32]; bb.y = l1[32];    // gate g
            acc2 = __builtin_amdgcn_wmma_f32_16x16x4_f32(false, a, false, bb, (short)0, acc2, false, false);
            bb.x = l0[48]; bb.y = l1[48];    // gate o
            acc3 = __builtin_amdgcn_wmma_f32_16x16x4_f32(false, a, false, bb, (short)0, acc3, false, false);
        }

        const int t_in = dir ? (T_DIM - 1 - s) : s;  // backward scans reversed xg
        const float* __restrict__ xgrow = xg + (size_t)t_in * B_DIM * G_DIM;
        float* __restrict__ orow = out + (size_t)s * B_DIM * 2 * H_DIM;

        for (int r = 0; r < 8; ++r) {
            int b = wave * 16 + r + mhi;
            const float* __restrict__ xgb = xgrow + (size_t)b * G_DIM;
            float gi = acc0[r] + xgb[hcol];
            float gf = acc1[r] + xgb[H_DIM + hcol];
            float gc = acc2[r] + xgb[2 * H_DIM + hcol];
            float go = acc3[r] + xgb[3 * H_DIM + hcol];
            float c = hsig(gf) * creg[r] + hsig(gi) * clip1(gc);
            creg[r] = c;
            orow[(size_t)b * 2 * H_DIM + (size_t)dir * H_DIM + hcol] = hsig(go) * clip1(c);
        }

        // Device-wide step barrier (monotonic arrival counter, agent scope)
        __syncthreads();
        if (tid == 0) {
            __threadfence();
            __hip_atomic_fetch_add(bar, 1u, __ATOMIC_RELEASE, __HIP_MEMORY_SCOPE_AGENT);
            const unsigned target = (unsigned)NBLOCKS_SEQ * (unsigned)(s + 1);
            while (__hip_atomic_load(bar, __ATOMIC_ACQUIRE, __HIP_MEMORY_SCOPE_AGENT) < target)
                __builtin_amdgcn_s_sleep(1);
        }
        __syncthreads();
    }

    // Emit hy (copy of out[T-1] in [2,B,H] layout) and cy (register cell state)
    float* __restrict__ hy = out + (size_t)T_DIM * B_DIM * 2 * H_DIM;
    float* __restrict__ cy = hy + (size_t)2 * B_DIM * H_DIM;
    const float* __restrict__ olast = out + (size_t)(T_DIM - 1) * B_DIM * 2 * H_DIM;
    for (int r = 0; r < 8; ++r) {
        int b = wave * 16 + r + mhi;
        size_t di = (size_t)dir * B_DIM * H_DIM + (size_t)b * H_DIM + hcol;
        hy[di] = olast[(size_t)b * 2 * H_DIM + (size_t)dir * H_DIM + hcol];
        cy[di] = creg[r];
    }
}

extern "C" void kernel_launch(void* const* d_in, const int* in_sizes, int n_in,
                              void* d_out, int out_size, void* d_ws, size_t ws_size,
                              hipStream_t stream) {
    (void)in_sizes; (void)n_in; (void)out_size; (void)ws_size;
    const float* x     = (const float*)d_in[0];
    const float* h0    = (const float*)d_in[1];
    const float* c0    = (const float*)d_in[2];
    const float* wih_f = (const float*)d_in[3];
    const float* whh_f = (const float*)d_in[4];
    const float* bih_f = (const float*)d_in[5];
    const float* bhh_f = (const float*)d_in[6];
    const float* wih_b = (const float*)d_in[7];
    const float* whh_b = (const float*)d_in[8];
    const float* bih_b = (const float*)d_in[9];
    const float* bhh_b = (const float*)d_in[10];
    float* out = (float*)d_out;

    // Workspace carve-up (256B aligned)
    char* ws = (char*)d_ws;
    size_t off = 0;
    auto carve = [&](size_t bytes) {
        void* p = ws + off;
        off += (bytes + 255) & ~(size_t)255;
        return p;
    };
    float* wihT_f = (float*)carve((size_t)IN_DIM * G_DIM * 4);
    float* wihT_b = (float*)carve((size_t)IN_DIM * G_DIM * 4);
    float* whhT_f = (float*)carve((size_t)H_DIM * G_DIM * 4);
    float* whhT_b = (float*)carve((size_t)H_DIM * G_DIM * 4);
    float* bias_f = (float*)carve((size_t)G_DIM * 4);
    float* bias_b = (float*)carve((size_t)G_DIM * 4);
    unsigned* bar = (unsigned*)carve(256);
    float* xg_f   = (float*)carve((size_t)T_DIM * B_DIM * G_DIM * 4);
    float* xg_b   = (float*)carve((size_t)T_DIM * B_DIM * G_DIM * 4);

    // 1) weight transpose + bias fold + barrier reset
    {
        int nthreads = G_DIM * IN_DIM;
        prep_kernel<<<(nthreads + 255) / 256, 256, 0, stream>>>(
            wih_f, whh_f, bih_f, bhh_f, wih_b, whh_b, bih_b, bhh_b,
            wihT_f, whhT_f, wihT_b, whhT_b, bias_f, bias_b, bar);
    }

    // 2) big input-projection GEMM (both directions)
    {
        const int NG = G_DIM / 64;
        const int MT = (T_DIM * B_DIM) / 16;
        int waves  = 2 * MT * NG;          // 131072 waves
        int blocks = waves / 8;            // 256 threads = 8 waves/block
        xg_gemm_kernel<<<blocks, 256, 0, stream>>>(
            x, wihT_f, wihT_b, bias_f, bias_b, xg_f, xg_b);
    }

    // 3) persistent recurrence: one launch, all 512 timesteps.
    //    LDS: 512 x 72 floats = 147456 bytes (fits 320KB/WGP, 2 blocks/WGP).
    {
        size_t shbytes = (size_t)H_DIM * PANEL_STRIDE * sizeof(float);
        lstm_seq_kernel<<<NBLOCKS_SEQ, 128, shbytes, stream>>>(
            h0, c0, whhT_f, whhT_b, xg_f, xg_b, out, bar);
    }
}